// GELU230_23648089932073
// MI455X (gfx1250) — compile-verified
//
#include <hip/hip_runtime.h>
#include <math.h>

typedef __attribute__((ext_vector_type(2))) float v2f;
typedef __attribute__((ext_vector_type(8))) float v8f;

#define BT    16384   // B*T = 4*4096
#define D     1024
#define NBUF  32
#define K1_BLOCKS 128
#define ROWS_PER_K1 (BT / K1_BLOCKS)   // 128

// tanh-GELU, branch-free: x * sigmoid(2*sqrt(2/pi)*(x + 0.044715*x^3))
// -> one v_exp_f32 + one v_rcp_f32, no exec-mask divergence.
__device__ __forceinline__ float gelu_f(float v) {
    float t = fmaf(0.044715f * v * v, v, v);          // x + 0.044715 x^3
    float z = 1.5957691216057308f * t;                // 2*sqrt(2/pi)*t
    float e = __expf(-z);                             // v_exp_f32 (scaled), no branches
    return v * __builtin_amdgcn_rcpf(1.0f + e);       // x * sigmoid(z)
}

// ---------------- Kernel 1: per-block column sums of gelu(x) ----------------
__global__ void k_colsum(const float* __restrict__ x, float* __restrict__ partial) {
    const int tid = threadIdx.x;          // 256 threads -> 4 cols each (float4)
    const int blk = blockIdx.x;           // 128 blocks -> 128 rows each
    const float4* base = reinterpret_cast<const float4*>(x + (size_t)blk * ROWS_PER_K1 * D) + tid;
    float4 acc = make_float4(0.f, 0.f, 0.f, 0.f);
    #pragma unroll 4
    for (int r = 0; r < ROWS_PER_K1; ++r) {
        float4 v = base[(size_t)r * (D / 4)];
        acc.x += gelu_f(v.x); acc.y += gelu_f(v.y);
        acc.z += gelu_f(v.z); acc.w += gelu_f(v.w);
    }
    reinterpret_cast<float4*>(partial + (size_t)blk * D)[tid] = acc;
}

// ---------------- Kernel 1b: deterministic reduce partials -> mean ----------
__global__ void k_mean(const float* __restrict__ partial, float* __restrict__ m) {
    const int col = blockIdx.x * 256 + threadIdx.x;   // 4 blocks x 256 = 1024
    float s = 0.f;
    #pragma unroll 8
    for (int p = 0; p < K1_BLOCKS; ++p) s += partial[(size_t)p * D + col];
    m[col] = s * (1.0f / (float)BT);
}

// ---------------- Kernel 2: cosine sims, argmax, write normalized ep --------
__global__ void k_select(const float* __restrict__ buf, const unsigned char* __restrict__ mask,
                         const float* __restrict__ m, float* __restrict__ ep) {
    __shared__ int   sIdx;
    __shared__ float sScale;
    const int tid = threadIdx.x;
    if (tid < NBUF) {                     // single wave: lane n handles buf row n
        const int n = tid;
        float dot = 0.f, nb2 = 0.f, nm2 = 0.f;
        for (int d = 0; d < D; ++d) {
            float bv = buf[(size_t)n * D + d];
            float mv = m[d];
            dot = fmaf(bv, mv, dot);
            nb2 = fmaf(bv, bv, nb2);
            nm2 = fmaf(mv, mv, nm2);
        }
        float sim = dot / (fmaxf(sqrtf(nb2), 1e-12f) * fmaxf(sqrtf(nm2), 1e-12f));
        if (!mask[n]) sim = -1.0f;
        // first-index argmax via wave reduction
        int   bi = n;
        float bv = sim;
        for (int off = 16; off >= 1; off >>= 1) {
            float ov = __shfl_down(bv, off, 32);
            int   oi = __shfl_down(bi, off, 32);
            if (ov > bv || (ov == bv && oi < bi)) { bv = ov; bi = oi; }
        }
        int widx = __shfl(bi, 0, 32);
        float nbw = __shfl(nb2, widx, 32);
        if (tid == 0) {
            sIdx = widx;
            sScale = 1.0f / fmaxf(sqrtf(nbw), 1e-12f);
        }
    }
    __syncthreads();
    const int   idx = sIdx;
    const float sc  = sScale;
    float4 v = reinterpret_cast<const float4*>(buf + (size_t)idx * D)[tid];
    float4 o = make_float4(v.x * sc, v.y * sc, v.z * sc, v.w * sc);
    reinterpret_cast<float4*>(ep)[tid] = o;
}

// ---------------- Kernel 3: fused gate (WMMA dot) + output ------------------
// 256 threads = 8 waves; each wave owns 16 rows. Grid = BT/128 = 128 blocks.
__global__ void __launch_bounds__(256)
k_gate_apply(const float* __restrict__ x, const float* __restrict__ ep,
             const float* __restrict__ p_log_tau, const float* __restrict__ p_log_blend,
             const float* __restrict__ p_log_beta_floor, const float* __restrict__ p_logit_theta_hi,
             float* __restrict__ out) {
    const int lane = threadIdx.x & 31;
    const int wave = threadIdx.x >> 5;
    const int rowBase = blockIdx.x * 128 + wave * 16;

    const float tau   = expf(p_log_tau[0]);
    const float alpha = 1.0f / (1.0f + expf(-p_log_blend[0]));
    const float beta  = fminf(fmaxf(expf(p_log_beta_floor[0]), 1.0f), 20.0f);
    const float theta = 1.0f / (1.0f + expf(-p_logit_theta_hi[0]));

    // A-matrix lane mapping for V_WMMA_F32_16X16X4_F32 (16x4 f32):
    // lanes 0-15: row = lane, regs {K=0, K=1}; lanes 16-31: row = lane-16, regs {K=2, K=3}
    const int rsel = lane & 15;
    const int koff = (lane >> 4) * 2;
    const float* myrow = x + (size_t)(rowBase + rsel) * D + koff;
    const float* epk   = ep + koff;

    v8f  acc = {0.f, 0.f, 0.f, 0.f, 0.f, 0.f, 0.f, 0.f};
    float ss = 0.f;
    #pragma unroll 8
    for (int c = 0; c < D; c += 4) {
        float2 xv = *reinterpret_cast<const float2*>(myrow + c);
        float2 ev = *reinterpret_cast<const float2*>(epk + c);
        float a0 = gelu_f(xv.x);
        float a1 = gelu_f(xv.y);
        v2f A = {a0, a1};
        v2f B = {ev.x, ev.y};   // B[k][j] = ep[c+k] for all j (broadcast columns)
        acc = __builtin_amdgcn_wmma_f32_16x16x4_f32(
            false, A, false, B, (short)0, acc, false, false);
        ss = fmaf(a0, a0, ss);
        ss = fmaf(a1, a1, ss);
    }
    // combine the two column-halves of each row's sum-of-squares
    ss += __shfl_xor(ss, 16, 32);

    // extract dot_row from the 16x16 f32 C/D layout:
    // VGPR r: lanes 0-15 hold D[r][N=lane], lanes 16-31 hold D[r+8][N=lane-16];
    // all columns equal -> component r of a lane in the right half gives dot.
    const int r7 = lane & 7;
    float comp = acc[0];
    comp = (r7 == 1) ? acc[1] : comp;
    comp = (r7 == 2) ? acc[2] : comp;
    comp = (r7 == 3) ? acc[3] : comp;
    comp = (r7 == 4) ? acc[4] : comp;
    comp = (r7 == 5) ? acc[5] : comp;
    comp = (r7 == 6) ? acc[6] : comp;
    comp = (r7 == 7) ? acc[7] : comp;
    float dot = __shfl(comp, (lane < 8) ? lane : lane + 8, 32);

    float tok  = dot / fmaxf(sqrtf(ss), 1e-12f);
    float gb   = (1.0f - alpha) + alpha * expf(-tau * tok);
    float fg   = 1.0f / (1.0f + expf(beta * (tok - theta)));   // sigmoid(-beta*(tok-theta))
    float gate = fminf(fmaxf(gb * fg, 0.02f), 1.05f);          // valid on lanes 0..15 (row = lane)

    // Phase B: out = gelu(x) * gate[row], coalesced float4 per row
    for (int r = 0; r < 16; ++r) {
        float g = __shfl(gate, r, 32);
        const float4* xp = reinterpret_cast<const float4*>(x   + (size_t)(rowBase + r) * D);
        float4*       op = reinterpret_cast<float4*>(      out + (size_t)(rowBase + r) * D);
        #pragma unroll
        for (int i = lane; i < D / 4; i += 32) {
            float4 v = xp[i];
            float4 o = make_float4(gelu_f(v.x) * g, gelu_f(v.y) * g,
                                   gelu_f(v.z) * g, gelu_f(v.w) * g);
            op[i] = o;
        }
    }
}

extern "C" void kernel_launch(void* const* d_in, const int* in_sizes, int n_in,
                              void* d_out, int out_size, void* d_ws, size_t ws_size,
                              hipStream_t stream) {
    const float*         x    = (const float*)d_in[0];          // [4,4096,1024]
    const float*         buf  = (const float*)d_in[1];          // [32,1024]
    const unsigned char* mask = (const unsigned char*)d_in[2];  // [32] bool
    const float*         lt   = (const float*)d_in[3];
    const float*         lb   = (const float*)d_in[4];
    const float*         lbf  = (const float*)d_in[5];
    const float*         lth  = (const float*)d_in[6];
    float*               out  = (float*)d_out;

    // workspace layout (floats): partials[128*1024] | m[1024] | ep[1024]
    float* partial = (float*)d_ws;
    float* m       = partial + (size_t)K1_BLOCKS * D;
    float* ep      = m + D;

    k_colsum    <<<K1_BLOCKS, 256, 0, stream>>>(x, partial);
    k_mean      <<<D / 256,   256, 0, stream>>>(partial, m);
    k_select    <<<1,         256, 0, stream>>>(buf, mask, m, ep);
    k_gate_apply<<<BT / 128,  256, 0, stream>>>(x, ep, lt, lb, lbf, lth, out);
}